// NonMaxSuppression_1288490189238
// MI455X (gfx1250) — compile-verified
//
#include <hip/hip_runtime.h>
#include <stdint.h>

// ---------------- problem constants (match reference) ----------------
#define B_      16
#define C_      80
#define N_      5000
#define MAXOUT  100
#define BC_     (B_ * C_)
#define THREADS 256
#define NEG_INF (-__builtin_inff())

// ext-vector + address-space typed pointers for the async-DMA builtin
// (clang's diagnostic showed param0 = 'int __vector(4) __device__ *')
typedef int v4i __attribute__((ext_vector_type(4)));
typedef __attribute__((address_space(1))) v4i* g_v4i_ptr;   // global
typedef __attribute__((address_space(3))) v4i* l_v4i_ptr;   // LDS

// ---------------------------------------------------------------------
// Kernel 1: per-(batch,class) greedy NMS, one workgroup per problem.
// Boxes + masked scores staged into LDS (CDNA5: 320KB/WGP makes this fit),
// boxes via async global->LDS DMA when the builtin is available.
// ---------------------------------------------------------------------
__global__ __launch_bounds__(THREADS) void nms_kernel(
    const float* __restrict__ boxes,   // [B, N, 4]
    const float* __restrict__ scores,  // [B, C, N]
    int* __restrict__ counts,          // [BC]
    int* __restrict__ sels)            // [BC, MAXOUT]
{
    __shared__ float4 sbox[N_];                 // 80,000 B
    __shared__ float  ssc[N_];                  // 20,000 B
    __shared__ unsigned long long wkey;         // block argmax key

    const int bc  = blockIdx.x;
    const int b   = bc / C_;
    const int tid = threadIdx.x;

    const float4* gbox = (const float4*)(boxes + (size_t)b * N_ * 4);
    const float*  gsc  = scores + (size_t)bc * N_;

#if defined(__gfx1250__) && \
    __has_builtin(__builtin_amdgcn_global_load_async_to_lds_b128) && \
    __has_builtin(__builtin_amdgcn_s_wait_asynccnt)
    // CDNA5 async DMA: global -> LDS without touching VGPRs (ASYNCcnt).
    for (int i = tid; i < N_; i += THREADS) {
        __builtin_amdgcn_global_load_async_to_lds_b128(
            (g_v4i_ptr)(gbox + i),
            (l_v4i_ptr)(&sbox[i]),
            0, 0);
    }
    // Scores need a transform (score-threshold mask), so stage them normally.
    for (int i = tid; i < N_; i += THREADS) {
        float s = gsc[i];
        ssc[i] = (s > 0.5f) ? s : NEG_INF;      // strict >, as in reference
    }
    __builtin_amdgcn_s_wait_asynccnt(0);        // this wave's DMAs landed
    __syncthreads();                            // everyone's DMAs landed
#else
    for (int i = tid; i < N_; i += THREADS) sbox[i] = gbox[i];
    for (int i = tid; i < N_; i += THREADS) {
        float s = gsc[i];
        ssc[i] = (s > 0.5f) ? s : NEG_INF;
    }
    __syncthreads();
#endif

    int   nsel = 0;
    int   prev = -1;
    float px1 = 0.f, py1 = 0.f, px2 = 0.f, py2 = 0.f, parea = 0.f;

    for (int k = 0; k < MAXOUT; ++k) {
        if (tid == 0) wkey = 0ull;
        __syncthreads();

        // One fused LDS sweep: apply previous winner's suppression, then
        // accumulate argmax among survivors. Self-suppression is implied
        // (IoU(sel,sel)=1 > 0.5).
        unsigned long long lk = 0ull;
        for (int i = tid; i < N_; i += THREADS) {
            float s = ssc[i];
            if (s == NEG_INF) continue;                 // inactive
            float4 bx = sbox[i];
            if (prev >= 0) {
                float ix1 = fmaxf(px1, bx.x);
                float iy1 = fmaxf(py1, bx.y);
                float ix2 = fminf(px2, bx.z);
                float iy2 = fminf(py2, bx.w);
                float iw  = fmaxf(ix2 - ix1, 0.0f);
                float ih  = fmaxf(iy2 - iy1, 0.0f);
                float inter = iw * ih;
                float area  = (bx.z - bx.x) * (bx.w - bx.y);
                // iou > 0.5  <=>  inter > 0.5*(area+parea-inter)
                //            <=>  3*inter > area + parea   (division-free)
                if (3.0f * inter > area + parea) { ssc[i] = NEG_INF; continue; }
            }
            // Monotonic key: positive-float bits || inverted index.
            // Max key == max score; ties -> smallest index (JAX argmax).
            unsigned long long key =
                ((unsigned long long)__float_as_uint(s) << 32) |
                (unsigned long long)(0xFFFFFFFFu - (unsigned)i);
            lk = (key > lk) ? key : lk;
        }

        // wave32 butterfly max-reduce (5 steps), then ds_max_u64 across waves.
        for (int off = 16; off > 0; off >>= 1) {
            unsigned long long o = __shfl_xor(lk, off, 32);
            lk = (o > lk) ? o : lk;
        }
        if ((tid & 31) == 0 && lk != 0ull) atomicMax(&wkey, lk);
        __syncthreads();
        unsigned long long w = wkey;
        __syncthreads();                      // all reads done before next reset

        if (w == 0ull) break;                 // uniform: no active boxes left
        int sel = (int)(0xFFFFFFFFu - (unsigned)(w & 0xFFFFFFFFu));
        if (tid == 0) sels[bc * MAXOUT + k] = sel;
        nsel = k + 1;

        float4 pb = sbox[sel];
        px1 = pb.x; py1 = pb.y; px2 = pb.z; py2 = pb.w;
        parea = (px2 - px1) * (py2 - py1);
        prev = sel;
    }

    if (tid == 0) counts[bc] = nsel;
}

// ---------------------------------------------------------------------
// Kernel 2: exclusive prefix-sum over counts[1280], single workgroup.
// Each thread owns 5 contiguous elements; wave32 shuffle scan across
// per-thread totals; cross-wave offsets through LDS.
// ---------------------------------------------------------------------
#define SEG 5   // 1280 / 256
__global__ __launch_bounds__(256) void scan_kernel(
    const int* __restrict__ counts, int* __restrict__ offsets)
{
    __shared__ int wsum[8];
    const int tid  = threadIdx.x;
    const int base = tid * SEG;

    int c[SEG];
    int tot = 0;
    for (int j = 0; j < SEG; ++j) { c[j] = counts[base + j]; tot += c[j]; }

    int incl = tot;                                 // wave inclusive scan
    for (int off = 1; off < 32; off <<= 1) {
        int v = __shfl_up(incl, off, 32);
        if ((tid & 31) >= off) incl += v;
    }
    const int wid = tid >> 5;
    if ((tid & 31) == 31) wsum[wid] = incl;
    __syncthreads();

    int wofs = 0;
    for (int j = 0; j < wid; ++j) wofs += wsum[j];  // 8 waves: tiny loop

    int excl = wofs + incl - tot;                   // exclusive for my segment
    for (int j = 0; j < SEG; ++j) { offsets[base + j] = excl; excl += c[j]; }
}

// ---------------------------------------------------------------------
// Kernel 3: zero the (zero-padded) output.
// ---------------------------------------------------------------------
__global__ void zero_kernel(int* __restrict__ out, int n)
{
    int i = blockIdx.x * blockDim.x + threadIdx.x;
    if (i < n) out[i] = 0;
}

// ---------------------------------------------------------------------
// Kernel 4: scatter compacted (b, c, box_idx) triples (global sequential
// packing, exactly like the reference's cumsum/put).
// ---------------------------------------------------------------------
__global__ __launch_bounds__(128) void scatter_kernel(
    const int* __restrict__ counts, const int* __restrict__ offsets,
    const int* __restrict__ sels, int* __restrict__ out)
{
    const int bc  = blockIdx.x;
    const int cnt = counts[bc];
    const int off = offsets[bc];
    const int b   = bc / C_;
    const int cl  = bc % C_;
    for (int k = threadIdx.x; k < cnt; k += 128) {
        int p = (off + k) * 3;
        out[p + 0] = b;
        out[p + 1] = cl;
        out[p + 2] = sels[bc * MAXOUT + k];
    }
}

// ---------------------------------------------------------------------
// Workspace layout (ints):
//   [0, BC)                    counts
//   [BC, 2*BC)                 offsets
//   [2*BC, 2*BC + BC*MAXOUT)   selections      => ~522 KB total
// ---------------------------------------------------------------------
extern "C" void kernel_launch(void* const* d_in, const int* in_sizes, int n_in,
                              void* d_out, int out_size, void* d_ws, size_t ws_size,
                              hipStream_t stream)
{
    const float* boxes  = (const float*)d_in[0];   // [16,5000,4] f32
    const float* scores = (const float*)d_in[1];   // [16,80,5000] f32
    int* out = (int*)d_out;                        // [BC*MAXOUT, 3] i32

    int* counts  = (int*)d_ws;
    int* offsets = counts + BC_;
    int* sels    = offsets + BC_;

    zero_kernel<<<(out_size + 255) / 256, 256, 0, stream>>>(out, out_size);
    nms_kernel<<<BC_, THREADS, 0, stream>>>(boxes, scores, counts, sels);
    scan_kernel<<<1, 256, 0, stream>>>(counts, offsets);
    scatter_kernel<<<BC_, 128, 0, stream>>>(counts, offsets, sels, out);
}